// CCModel_8332236554522
// MI455X (gfx1250) — compile-verified
//
#include <hip/hip_runtime.h>

#define NC_    8192
#define NT_    4096
#define NLAGC  51
#define LOUT   103          // output lags
#define LP     112          // padded lag stride (7 * 16)
#define PADR   16           // zero pad rows (top & bottom) for WMMA band matmul
#define WSROWS (NC_ + 2 * PADR)
#define DT_F   0.01f

typedef float v2f  __attribute__((ext_vector_type(2)));
typedef float v8f  __attribute__((ext_vector_type(8)));
typedef unsigned int u32x4 __attribute__((ext_vector_type(4)));
typedef int  i32x4 __attribute__((ext_vector_type(4)));
typedef int  i32x8 __attribute__((ext_vector_type(8)));

// ---------------------------------------------------------------------------
// TDM: DMA one contiguous row of f32 from global memory into LDS.
// D# built per CDNA5 ISA §8 (group0: count=1,lds_addr,global_addr,type=2;
// group1: data_size=4B, tensor_dim0=tile_dim0=nelts, 1-D tile).
// This toolchain (clang-23 / therock-10.0) uses the 6-arg builtin form.
// ---------------------------------------------------------------------------
__device__ __forceinline__ void tdm_load_row_f32(unsigned lds_byte_off,
                                                 const float* gsrc,
                                                 unsigned nelts) {
  unsigned long long ga = (unsigned long long)gsrc;
  u32x4 g0;
  g0[0] = 1u;                                              // count=1, user mode
  g0[1] = lds_byte_off;                                    // lds_addr (bytes)
  g0[2] = (unsigned)(ga & 0xffffffffull);                  // global_addr[31:0]
  g0[3] = ((unsigned)(ga >> 32) & 0x01ffffffu) | (2u << 30); // addr[56:32], type=2
  i32x8 g1;
  g1[0] = (int)(2u << 16);                                 // data_size = 4 bytes
  g1[1] = (int)((nelts & 0xffffu) << 16);                  // tensor_dim0[15:0]
  g1[2] = (int)(((nelts >> 16) & 0xffffu) | (1u << 16));   // dim0[31:16], tensor_dim1=1
  g1[3] = (int)((nelts & 0xffffu) << 16);                  // tile_dim0
  g1[4] = 0;                                               // tile_dim1=0 (1-D), tile_dim2=0
  g1[5] = (int)nelts;                                      // tensor_dim0_stride lo
  g1[6] = 0;
  g1[7] = 0;
  i32x4 z4 = {0, 0, 0, 0};
  i32x8 z8 = {0, 0, 0, 0, 0, 0, 0, 0};
  __builtin_amdgcn_tensor_load_to_lds(g0, g1, z4, z4, z8, 0);
}

// ---------------------------------------------------------------------------
// Zero the 16 pad rows above/below the raw-xcorr workspace (WSROWS x LP).
// ---------------------------------------------------------------------------
__global__ __launch_bounds__(256) void ws_init_kernel(float* __restrict__ raw) {
  int i = blockIdx.x * 256 + threadIdx.x;
  if (i < PADR * LP) {
    raw[i] = 0.0f;
    raw[(size_t)(NC_ + PADR) * LP + i] = 0.0f;
  }
}

// ---------------------------------------------------------------------------
// Stage A: per-channel cross-correlation (raw, unsmoothed).
// One block per channel. TDM loads both rows into LDS; each thread owns a
// (lag-quad q, time-slice s) pair: 4 sliding-window lags over 512 samples
// => 2 LDS reads per 4 FMAs. 8 slices reduced with wave shuffles.
// ---------------------------------------------------------------------------
__global__ __launch_bounds__(256) void xcorr_raw_kernel(const float* __restrict__ d1,
                                                        const float* __restrict__ d2,
                                                        float* __restrict__ raw) {
  __shared__ float s_x1[NT_ + 128];   // data1 at offset 64, 64 zeros each side
  __shared__ float s_x2[NT_];
  const int c   = blockIdx.x;
  const int tid = threadIdx.x;

  if (tid < 64) {                     // zero the halo regions
    s_x1[tid] = 0.0f;
    s_x1[NT_ + 64 + tid] = 0.0f;
  }
  if (tid < 32) {                     // wave 0 drives the Tensor Data Mover
    tdm_load_row_f32((unsigned)(unsigned long long)(void*)&s_x1[64],
                     d1 + (size_t)c * NT_, NT_);
    tdm_load_row_f32((unsigned)(unsigned long long)(void*)&s_x2[0],
                     d2 + (size_t)c * NT_, NT_);
    __builtin_amdgcn_s_wait_tensorcnt(0);
  }
  __syncthreads();

  const int q = tid >> 3;             // lag quad: lags 4q..4q+3   (0..25 used)
  const int s = tid & 7;              // time slice: t in [512s, 512s+512)
  float a0 = 0.0f, a1 = 0.0f, a2 = 0.0f, a3 = 0.0f;
  if (q < 26) {
    // d1[c, t + l - 51] lives at s_x1[64 + t + l - 51] = s_x1[13 + 4q + t + j]
    const float* x1 = &s_x1[13 + 4 * q + 512 * s];
    const float* x2 = &s_x2[512 * s];
    float c0 = x1[0], c1 = x1[1], c2 = x1[2];
    #pragma unroll 8
    for (int t = 0; t < 512; ++t) {
      float b  = x2[t];
      float c3 = x1[t + 3];
      a0 = fmaf(c0, b, a0);
      a1 = fmaf(c1, b, a1);
      a2 = fmaf(c2, b, a2);
      a3 = fmaf(c3, b, a3);
      c0 = c1; c1 = c2; c2 = c3;
    }
  }
  // reduce the 8 slices (lanes differing in bits 0..2 of the lane id)
  for (int m = 1; m <= 4; m <<= 1) {
    a0 += __shfl_xor(a0, m, 32);
    a1 += __shfl_xor(a1, m, 32);
    a2 += __shfl_xor(a2, m, 32);
    a3 += __shfl_xor(a3, m, 32);
  }
  float* row = raw + (size_t)(c + PADR) * LP;
  if (s == 0 && q < 26) {
    int l = 4 * q;
    row[l]     = a0;
    row[l + 1] = a1;
    row[l + 2] = a2;
    if (l + 3 < LOUT) row[l + 3] = a3;
  }
  if (tid >= 208 && tid < 208 + (LP - LOUT))   // zero padded lag columns
    row[LOUT + (tid - 208)] = 0.0f;
}

// ---------------------------------------------------------------------------
// Stage B: channel moving-average as a banded f32 matmul on the WMMA unit.
// Smoothed[c0+m, l0+n] = sum_j W[m][j] * raw_ws[c0+j, l0+n],  j in [0,48),
// W[m][j] = 0.05 iff (j-m) in [6,25]  (== mean over channels [c-10, c+9] /20,
// boundaries handled by the zero pad rows). One wave per 16x16 output tile.
// ---------------------------------------------------------------------------
__global__ __launch_bounds__(32) void smooth_wmma_kernel(const float* __restrict__ raw,
                                                         float* __restrict__ xcor_out) {
  const int cb   = blockIdx.x / (LP / 16);
  const int lt   = blockIdx.x % (LP / 16);
  const int c0   = cb * 16;
  const int l0   = lt * 16;
  const int lane = threadIdx.x;
  const int m    = lane & 15;     // A row / B-C-D column index
  const int half = lane >> 4;

  v8f acc = {};
  // nonzero band is j in [6,40]; K-tiles k0=4..40 cover it
  for (int k0 = 4; k0 <= 40; k0 += 4) {
    const int kb = k0 + half * 2;
    v2f a, b;
    a.x = ((kb     - m) >= 6 && (kb     - m) <= 25) ? 0.05f : 0.0f;
    a.y = ((kb + 1 - m) >= 6 && (kb + 1 - m) <= 25) ? 0.05f : 0.0f;
    const float* r0 = raw + (size_t)(c0 + kb) * LP + l0 + m;
    b.x = r0[0];
    b.y = r0[LP];
    acc = __builtin_amdgcn_wmma_f32_16x16x4_f32(false, a, false, b,
                                                (short)0, acc, false, false);
  }
  const int l = l0 + m;
  if (l < LOUT) {
    #pragma unroll
    for (int j = 0; j < 8; ++j) {
      int ch = c0 + j + half * 8;   // C/D layout: VGPR j -> M=j (+8 for hi half)
      xcor_out[(size_t)ch * LOUT + l] = acc[j];
    }
  }
}

// ---------------------------------------------------------------------------
// Stage C: per-channel picks on the smoothed correlation.
// ---------------------------------------------------------------------------
__global__ __launch_bounds__(256) void pick_kernel(const float* __restrict__ xcor,
                                                   float* __restrict__ vmain_out,
                                                   float* __restrict__ vside_out,
                                                   float* __restrict__ tmax_out) {
  int c = blockIdx.x * 256 + threadIdx.x;
  if (c >= NC_) return;
  const float* R = xcor + (size_t)c * LOUT;
  float best = -1.0f;
  int   imax = 0;
  float vpos = -3.402823466e38f, vneg = 3.402823466e38f;
  for (int l = 0; l < LOUT; ++l) {
    float v  = R[l];
    float av = fabsf(v);
    if (av > best) { best = av; imax = l; }   // strict > == first-occurrence argmax
    vpos = fmaxf(vpos, v);
    vneg = fminf(vneg, v);
  }
  float vmain = R[imax];
  vmain_out[c] = vmain;
  vside_out[c] = (vmain >= 0.0f) ? vneg : vpos;
  tmax_out[c]  = (float)(imax - NLAGC) * DT_F;
}

// ---------------------------------------------------------------------------
extern "C" void kernel_launch(void* const* d_in, const int* in_sizes, int n_in,
                              void* d_out, int out_size, void* d_ws, size_t ws_size,
                              hipStream_t stream) {
  const float* d1 = (const float*)d_in[0];
  const float* d2 = (const float*)d_in[1];
  // d_in[2]/d_in[3] (event1/event2) are unused by the reference.

  float* out  = (float*)d_out;
  float* xcor = out;                              // (1, NC, 103) smoothed
  float* vmax = out + (size_t)NC_ * LOUT;         // vmain
  float* vmin = vmax + NC_;                       // vside
  float* tmax = vmin + NC_;
  float* raw  = (float*)d_ws;                     // WSROWS x LP raw xcorr

  ws_init_kernel<<<(PADR * LP + 255) / 256, 256, 0, stream>>>(raw);
  xcorr_raw_kernel<<<NC_, 256, 0, stream>>>(d1, d2, raw);
  smooth_wmma_kernel<<<(NC_ / 16) * (LP / 16), 32, 0, stream>>>(raw, xcor);
  pick_kernel<<<(NC_ + 255) / 256, 256, 0, stream>>>(xcor, vmax, vmin, tmax);
}